// CWICLinear_52553219834139
// MI455X (gfx1250) — compile-verified
//
#include <hip/hip_runtime.h>

typedef float v2f __attribute__((ext_vector_type(2)));
typedef float v8f __attribute__((ext_vector_type(8)));

#define IN_F   2048
#define OUT_F  8192
#define TOK    2048
#define NSTR   16
#define STRIPE 512

#define M_TILE  128
#define N_TILE  64
#define K_CHUNK 16
#define THREADS 256

// sqrt(2048) = threshold_lr_scale * sqrt(in_features)
#define THR_SCALE 45.25483399593904156165403917471f

// ---------------------------------------------------------------------------
// Init kernel: dense_params is a constant, active_params accumulates atomics.
// ---------------------------------------------------------------------------
__global__ void cwic_init_stats(float* __restrict__ dense,
                                float* __restrict__ active) {
    int t = blockIdx.x * blockDim.x + threadIdx.x;
    if (t < TOK) {
        dense[t]  = 16777216.0f;   // STRIPE * N_STRIPES * IN_F = 512*16*2048
        active[t] = 0.0f;
    }
}

// ---------------------------------------------------------------------------
// Main kernel: per-stripe gated GEMM using V_WMMA_F32_16X16X4_F32.
//   y[t, s*512+o] = sum_i w[i, s*512+o] * (|x-med|>thr ? x : med) + bias
// Grid: (STRIPE/N_TILE, TOK/M_TILE, NSTR), block: 256 threads (8 waves).
// ---------------------------------------------------------------------------
__global__ __launch_bounds__(THREADS)
void cwic_gemm(const float* __restrict__ x,     // [TOK, IN_F]
               const float* __restrict__ w,     // [IN_F, OUT_F]
               const float* __restrict__ bias,  // [OUT_F]
               const float* __restrict__ thrs,  // [NSTR, IN_F]
               const float* __restrict__ med,   // [IN_F]
               const float* __restrict__ stdv,  // [IN_F]
               float* __restrict__ y,           // [TOK, OUT_F]
               float* __restrict__ active)      // [TOK]
{
    __shared__ float As[M_TILE][K_CHUNK + 1];  // +1 pad: conflict-free frag reads
    __shared__ float Bs[K_CHUNK][N_TILE + 1];

    const int tid   = threadIdx.x;
    const int lane  = tid & 31;
    const int wave  = tid >> 5;                 // 0..7 -> M sub-row
    const int nBase = blockIdx.x * N_TILE;      // within stripe [0,512)
    const int mBase = blockIdx.y * M_TILE;      // token base
    const int s     = blockIdx.z;               // stripe

    const v8f vzero = {0.f, 0.f, 0.f, 0.f, 0.f, 0.f, 0.f, 0.f};
    v8f acc[4];
#pragma unroll
    for (int i = 0; i < 4; ++i) acc[i] = vzero;

    // A-loader geometry: thread handles column (tid&15), rows (tid>>4)+16j
    const int aRow0 = tid >> 4;   // 0..15
    const int aCol  = tid & 15;   // 0..15
    int cnt[8] = {0, 0, 0, 0, 0, 0, 0, 0};

    const int laneM = lane & 15;          // M (A) / N (B) index within fragment
    const int kHalf = (lane >> 4) * 2;    // K pair selector per half-wave

    for (int kBase = 0; kBase < IN_F; kBase += K_CHUNK) {
        // ---- stage gated A tile: 128 x 16 -------------------------------
        {
            const int   i  = kBase + aCol;
            const float mi = med[i];
            const float ti = thrs[s * IN_F + i] * stdv[i] * THR_SCALE;
#pragma unroll
            for (int j = 0; j < 8; ++j) {
                const int   r    = aRow0 + 16 * j;
                const float xv   = x[(size_t)(mBase + r) * IN_F + i];
                const bool  gate = fabsf(xv - mi) > ti;
                As[r][aCol] = gate ? xv : mi;   // folds post_mu into the GEMM
                cnt[j] += gate ? 1 : 0;
            }
        }
        // ---- stage B tile: 16 x 64 --------------------------------------
#pragma unroll
        for (int j = 0; j < (K_CHUNK * N_TILE) / THREADS; ++j) {
            const int e  = tid + j * THREADS;
            const int kr = e >> 6;      // 0..15
            const int nc = e & 63;      // 0..63
            Bs[kr][nc] = w[(size_t)(kBase + kr) * OUT_F + s * STRIPE + nBase + nc];
        }
        // prefetch next K-chunk of x into L2 (global_prefetch_b8)
        if (kBase + K_CHUNK < IN_F)
            __builtin_prefetch(&x[(size_t)(mBase + aRow0) * IN_F + kBase + K_CHUNK + aCol], 0, 1);
        __syncthreads();

        // ---- 4 K-steps of V_WMMA_F32_16X16X4_F32 ------------------------
#pragma unroll
        for (int kk = 0; kk < 4; ++kk) {
            const int k0 = kk * 4 + kHalf;  // ISA A-layout: v0/v1 = K pair per half-wave
            v2f a;
            a.x = As[wave * 16 + laneM][k0];
            a.y = As[wave * 16 + laneM][k0 + 1];
#pragma unroll
            for (int nt = 0; nt < 4; ++nt) {
                v2f b;
                b.x = Bs[k0][nt * 16 + laneM];
                b.y = Bs[k0 + 1][nt * 16 + laneM];
                acc[nt] = __builtin_amdgcn_wmma_f32_16x16x4_f32(
                    /*neg_a=*/false, a, /*neg_b=*/false, b,
                    /*c_mod=*/(short)0, acc[nt],
                    /*reuse_a=*/false, /*reuse_b=*/false);
            }
        }
        __syncthreads();
    }

    // ---- write-out: C layout VGPR r -> M = r + 8*(lane>=16), N = lane&15 ----
    const int mRow = mBase + wave * 16 + (lane >> 4) * 8;
#pragma unroll
    for (int nt = 0; nt < 4; ++nt) {
        const int   o  = s * STRIPE + nBase + nt * 16 + laneM;
        const float bo = bias[o];
#pragma unroll
        for (int r = 0; r < 8; ++r)
            y[(size_t)(mRow + r) * OUT_F + o] = acc[nt][r] + bo;
    }

    // ---- active_params: exact fp32 integer accumulation (<= 2^24) ----------
    if (blockIdx.x == 0) {
#pragma unroll
        for (int j = 0; j < 8; ++j)
            atomicAdd(&active[mBase + aRow0 + 16 * j], 512.0f * (float)cnt[j]);
    }
}

// ---------------------------------------------------------------------------
extern "C" void kernel_launch(void* const* d_in, const int* in_sizes, int n_in,
                              void* d_out, int out_size, void* d_ws, size_t ws_size,
                              hipStream_t stream) {
    (void)in_sizes; (void)n_in; (void)out_size; (void)d_ws; (void)ws_size;

    const float* x    = (const float*)d_in[0];  // [1, 2048, 2048]
    const float* w    = (const float*)d_in[1];  // [2048, 8192]
    const float* bias = (const float*)d_in[2];  // [8192]
    const float* thrs = (const float*)d_in[3];  // [16, 2048]
    const float* med  = (const float*)d_in[4];  // [2048]
    const float* stdv = (const float*)d_in[5];  // [2048]

    float* y      = (float*)d_out;                    // 2048*8192 floats
    float* dense  = y + (size_t)TOK * OUT_F;          // 2048 floats
    float* active = dense + TOK;                      // 2048 floats

    cwic_init_stats<<<(TOK + 255) / 256, 256, 0, stream>>>(dense, active);

    dim3 grid(STRIPE / N_TILE, TOK / M_TILE, NSTR);   // (8, 16, 16)
    cwic_gemm<<<grid, THREADS, 0, stream>>>(x, w, bias, thrs, med, stdv, y, active);
}